// GCMCModel_78700980732450
// MI455X (gfx1250) — compile-verified
//
#include <hip/hip_runtime.h>
#include <hip/hip_bf16.h>

// ---------------------------------------------------------------------------
// GCMC model on MI455X (gfx1250), wave32.
// GEMMs (K=64) use V_WMMA_F32_16X16X4_F32 (full-precision f32 WMMA).
// Edge phase: wave-per-edge, float2 gathers + global_atomic_add_f32 scatters;
// tables (20.5 MB) are L2-resident (192 MB L2).
// ---------------------------------------------------------------------------

typedef __attribute__((ext_vector_type(2))) float v2f;
typedef __attribute__((ext_vector_type(8))) float v8f;

constexpr int NU = 50000;
constexpr int NI = 30000;
constexpr int KD = 64;          // embedding dim
constexpr int RR = 5;           // relations
constexpr int SS = 2;
constexpr int EE = 1000000;     // edges per relation
constexpr int BB = 8192;        // batch
constexpr int NN = NU + NI;     // 80000 nodes

#define CDIV(a, b) (((a) + (b) - 1) / (b))

// ---------------------------------------------------------------------------
// C[M x 64] = A[M x 64] @ W[64 x 64]   (W transposed if TRANSB)
// A rows < splitRows come from A0, the rest from A1 (fused concat[Gu;Gi]).
// One wave computes one 16x16 tile of C via 16 steps of wmma_f32_16x16x4_f32.
//
// VGPR layouts (ISA 7.12.2):
//   A 16x4 f32 : lane&15 = row M; (lane>>4)*2 selects K-pair {0,1} or {2,3}
//   B  4x16 f32: lane&15 = col N; (lane>>4)*2 selects K-row pair
//   C/D 16x16  : VGPR v -> row v + 8*(lane>>4), col lane&15
// ---------------------------------------------------------------------------
template <bool TRANSB>
__global__ __launch_bounds__(256) void gemm_k64_wmma(
    const float* __restrict__ A0, const float* __restrict__ A1, int splitRows,
    const float* __restrict__ W, float* __restrict__ C, int M)
{
    const int lane  = threadIdx.x & 31;
    const int gwave = blockIdx.x * 8 + (threadIdx.x >> 5);
    const int mt    = gwave >> 2;       // 4 N-tiles (64/16)
    const int nt    = gwave & 3;
    if (mt * 16 >= M) return;           // tile-uniform: EXEC all-ones inside

    const int row    = mt * 16 + (lane & 15);
    const int col    = nt * 16 + (lane & 15);
    const int kpair  = (lane >> 4) * 2; // 0 or 2

    const float* Arow = (row < splitRows) ? (A0 + (size_t)row * KD)
                                          : (A1 + (size_t)(row - splitRows) * KD);

    v8f acc = {0.f, 0.f, 0.f, 0.f, 0.f, 0.f, 0.f, 0.f};
#pragma unroll
    for (int k0 = 0; k0 < KD; k0 += 4) {
        const int kr = k0 + kpair;
        v2f a, b;
        a.x = Arow[kr];
        a.y = Arow[kr + 1];
        if (TRANSB) {                   // B[k][n] = W[n][k]
            b.x = W[col * KD + kr];
            b.y = W[col * KD + kr + 1];
        } else {
            b.x = W[kr * KD + col];
            b.y = W[(kr + 1) * KD + col];
        }
        acc = __builtin_amdgcn_wmma_f32_16x16x4_f32(
            false, a, false, b, (short)0, acc, false, false);
    }

    const int rbase = mt * 16 + ((lane >> 4) * 8);
#pragma unroll
    for (int v = 0; v < 8; ++v)
        C[(size_t)(rbase + v) * KD + col] = acc[v];
}

// ---------------------------------------------------------------------------
__global__ void fill0_kernel(float* __restrict__ p, int n)
{
    int i = blockIdx.x * blockDim.x + threadIdx.x;
    if (i < n) p[i] = 0.0f;
}

__global__ void edge_deg_kernel(const int* __restrict__ dst, float* __restrict__ deg)
{
    int e = blockIdx.x * blockDim.x + threadIdx.x;
    if (e < EE) atomicAdd(&deg[dst[e]], 1.0f);
}

__global__ void dinv_kernel(const float* __restrict__ deg, float* __restrict__ dinv)
{
    int i = blockIdx.x * blockDim.x + threadIdx.x;
    if (i < NN) {
        float d = deg[i];
        dinv[i] = (d > 0.0f) ? rsqrtf(d) : 0.0f;   // deg integral => max(deg,1)==deg
    }
}

// wave-per-edge: 32 lanes cover 64 floats as float2; atomic scatter into acc[dst]
__global__ __launch_bounds__(256) void edge_msg_kernel(
    const int* __restrict__ src, const int* __restrict__ dst,
    const float* __restrict__ dinv, const float* __restrict__ h,
    float* __restrict__ acc)
{
    int t    = blockIdx.x * 256 + threadIdx.x;
    int e    = t >> 5;
    int lane = t & 31;
    if (e >= EE) return;
    int s = src[e], d = dst[e];
    float norm = dinv[s] * dinv[d];
    const float2 v = ((const float2*)(h + (size_t)s * KD))[lane];
    float* ad = acc + (size_t)d * KD + lane * 2;
    atomicAdd(ad,     v.x * norm);
    atomicAdd(ad + 1, v.y * norm);
}

__global__ void reluacc_kernel(const float* __restrict__ acc, float* __restrict__ emb, int n)
{
    int i = blockIdx.x * blockDim.x + threadIdx.x;
    if (i < n) emb[i] += fmaxf(acc[i], 0.0f);
}

__global__ void gather_kernel(const float* __restrict__ z,
                              const int* __restrict__ user, const int* __restrict__ item,
                              float* __restrict__ zu, float* __restrict__ zi)
{
    int t = blockIdx.x * blockDim.x + threadIdx.x;
    if (t >= BB * KD) return;
    int b = t >> 6, k = t & 63;
    zu[t] = z[(size_t)user[b] * KD + k];
    zi[t] = z[(size_t)(NU + item[b]) * KD + k];
}

__global__ void qmix_kernel(const float* __restrict__ A, const float* __restrict__ P,
                            float* __restrict__ Q)
{
    int t = blockIdx.x * blockDim.x + threadIdx.x;
    if (t >= RR * KD * KD) return;
    int r = t / (KD * KD);
    int ij = t - r * (KD * KD);
    float s = 0.0f;
    for (int q = 0; q < SS; ++q) s += A[r * SS + q] * P[q * KD * KD + ij];
    Q[t] = s;
}

// pui[b*R + r] = dot(U[b], T[b]); wave-per-b with wave32 shuffle reduce
__global__ __launch_bounds__(256) void rowdot_kernel(
    const float* __restrict__ U, const float* __restrict__ T,
    float* __restrict__ pui, int r)
{
    int t    = blockIdx.x * 256 + threadIdx.x;
    int b    = t >> 5;
    int lane = t & 31;
    if (b >= BB) return;
    const float* u  = U + (size_t)b * KD;
    const float* tt = T + (size_t)b * KD;
    float s = u[lane] * tt[lane] + u[lane + 32] * tt[lane + 32];
#pragma unroll
    for (int off = 16; off > 0; off >>= 1) s += __shfl_down(s, off, 32);
    if (lane == 0) pui[b * RR + r] = s;
}

__global__ void softmax_kernel(const float* __restrict__ pui,
                               const float* __restrict__ rel,
                               float* __restrict__ xui)
{
    int b = blockIdx.x * blockDim.x + threadIdx.x;
    if (b >= BB) return;
    float p[RR];
    float m = -3.0e38f;
    for (int r = 0; r < RR; ++r) { p[r] = pui[b * RR + r]; m = fmaxf(m, p[r]); }
    float den = 0.0f, num = 0.0f;
    for (int r = 0; r < RR; ++r) {
        float e = expf(p[r] - m);
        den += e;
        num += rel[r] * e;
    }
    xui[b] = num / den;
}

// ---------------------------------------------------------------------------
extern "C" void kernel_launch(void* const* d_in, const int* in_sizes, int n_in,
                              void* d_out, int out_size, void* d_ws, size_t ws_size,
                              hipStream_t stream)
{
    const float* Gu  = (const float*)d_in[0];   // NU x 64
    const float* Gi  = (const float*)d_in[1];   // NI x 64
    const float* Wc  = (const float*)d_in[2];   // R x 64 x 64
    const float* Wd  = (const float*)d_in[3];   // 64 x 64
    const float* P   = (const float*)d_in[4];   // S x 64 x 64
    const float* A   = (const float*)d_in[5];   // R x S
    const float* rel = (const float*)d_in[6];   // R
    const int* esrc  = (const int*)d_in[7];     // R x E
    const int* edst  = (const int*)d_in[8];     // R x E
    const int* user  = (const int*)d_in[9];     // B
    const int* item  = (const int*)d_in[10];    // B

    float* out = (float*)d_out;                 // [0,B)=xui ; [B, B+B*R)=pui
    float* xui = out;
    float* pui = out + BB;

    // workspace layout (floats), ~68.5 MB total
    float* ws   = (float*)d_ws;
    float* emb  = ws;                   // N*K
    float* h    = emb  + (size_t)NN * KD;   // N*K  (reused as z)
    float* acc  = h    + (size_t)NN * KD;   // N*K
    float* deg  = acc  + (size_t)NN * KD;   // N
    float* dinv = deg  + NN;                // N
    float* zu   = dinv + NN;                // B*K
    float* zi   = zu   + (size_t)BB * KD;   // B*K
    float* Qb   = zi   + (size_t)BB * KD;   // R*K*K
    float* Tb   = Qb   + RR * KD * KD;      // B*K

    const int NK = NN * KD;
    const int gemmN_blocks = CDIV((NN / 16) * 4, 8);   // 2500
    const int gemmB_blocks = CDIV((BB / 16) * 4, 8);   // 256

    // emb = 0
    fill0_kernel<<<CDIV(NK, 256), 256, 0, stream>>>(emb, NK);

    // ---- GCN layer: emb = sum_r relu( segsum( norm * (x0 @ Wc[r])[src] ) ) ----
    for (int r = 0; r < RR; ++r) {
        const int* sr = esrc + (size_t)r * EE;
        const int* dr = edst + (size_t)r * EE;

        fill0_kernel<<<CDIV(NN, 256), 256, 0, stream>>>(deg, NN);
        edge_deg_kernel<<<CDIV(EE, 256), 256, 0, stream>>>(dr, deg);
        dinv_kernel<<<CDIV(NN, 256), 256, 0, stream>>>(deg, dinv);

        // h = [Gu;Gi] @ Wc[r]   (concat fused into A-operand load)
        gemm_k64_wmma<false><<<gemmN_blocks, 256, 0, stream>>>(
            Gu, Gi, NU, Wc + (size_t)r * KD * KD, h, NN);

        fill0_kernel<<<CDIV(NK, 256), 256, 0, stream>>>(acc, NK);
        edge_msg_kernel<<<CDIV(EE * 32, 256), 256, 0, stream>>>(sr, dr, dinv, h, acc);

        reluacc_kernel<<<CDIV(NK, 256), 256, 0, stream>>>(acc, emb, NK);
    }

    // ---- z = emb @ Wd (stored into h) ----
    gemm_k64_wmma<false><<<gemmN_blocks, 256, 0, stream>>>(emb, emb, NN, Wd, h, NN);

    // ---- decoder ----
    gather_kernel<<<CDIV(BB * KD, 256), 256, 0, stream>>>(h, user, item, zu, zi);
    qmix_kernel<<<CDIV(RR * KD * KD, 256), 256, 0, stream>>>(A, P, Qb);

    for (int r = 0; r < RR; ++r) {
        // T = zeta_i @ Q[r]^T   => T[b,i] = sum_j Q[r,i,j] * zi[b,j]
        gemm_k64_wmma<true><<<gemmB_blocks, 256, 0, stream>>>(
            zi, zi, BB, Qb + (size_t)r * KD * KD, Tb, BB);
        rowdot_kernel<<<CDIV(BB * 32, 256), 256, 0, stream>>>(zu, Tb, pui, r);
    }

    softmax_kernel<<<CDIV(BB, 256), 256, 0, stream>>>(pui, rel, xui);
}